// GAT_3453153706310
// MI455X (gfx1250) — compile-verified
//
#include <hip/hip_runtime.h>
#include <math.h>

#define F_IN   128
#define HID    64
#define NG     512
#define NCLS   2
#define NEG_SLOPE 0.2f

typedef __attribute__((ext_vector_type(2))) float    v2f;
typedef __attribute__((ext_vector_type(8))) float    v8f;
typedef __attribute__((ext_vector_type(4))) unsigned u32x4;
typedef __attribute__((ext_vector_type(8))) int      i32x8;
typedef __attribute__((ext_vector_type(4))) int      i32x4;

// ---------- helpers ----------
// Order-preserving float -> uint encoding so segment-max can use u32 atomics
__device__ __forceinline__ unsigned fenc(float f) {
  unsigned u = __float_as_uint(f);
  return (u & 0x80000000u) ? ~u : (u | 0x80000000u);
}
__device__ __forceinline__ float fdec(unsigned e) {
  unsigned u = (e & 0x80000000u) ? (e & 0x7FFFFFFFu) : ~e;
  return __uint_as_float(u);
}
#define ENC_NEG_INF 0x007FFFFFu  // fenc(-inf)

__device__ __forceinline__ float lrelu(float v) { return v > 0.f ? v : NEG_SLOPE * v; }

__device__ __forceinline__ void atomAddF(float* p, float v) {
  __hip_atomic_fetch_add(p, v, __ATOMIC_RELAXED, __HIP_MEMORY_SCOPE_AGENT);
}

// Low 32 bits of a generic pointer to a __shared__ object == LDS byte offset
// (flat-LDS aperture mapping uses the low 32 bits, ISA 10.2).
__device__ __forceinline__ unsigned lds_off(const void* p) {
  return (unsigned)(unsigned long long)(uintptr_t)p;
}

// ---------- Tensor Data Mover: 2-D fp32 tile load Global -> LDS ----------
// Packs D# group0/group1 per CDNA5 ISA §8.3/8.4 and issues TENSOR_LOAD_TO_LDS.
// dim0   : tensor row length in elements (contiguous)
// dim1   : tensor row count (rows >= dim1 are zero-filled by hardware)
// tile0/1: tile extent in elements / rows
// stride0: elements between consecutive rows
__device__ __forceinline__ void tdm_load_2d_f32(unsigned lds_byte_addr, const void* gaddr,
                                                unsigned dim0, unsigned dim1,
                                                unsigned tile0, unsigned tile1,
                                                unsigned stride0) {
  unsigned long long ga = (unsigned long long)(uintptr_t)gaddr;
  u32x4 g0;
  g0.x = 1u;                                                  // count=1, user descriptor
  g0.y = lds_byte_addr;                                       // lds_addr
  g0.z = (unsigned)(ga & 0xFFFFFFFFu);                        // global_addr[31:0]
  g0.w = (unsigned)((ga >> 32) & 0x01FFFFFFu) | (2u << 30);   // global_addr[56:32] | type=2
  i32x8 g1;
  g1[0] = (int)(2u << 16);                                    // data_size=4B; mask/flags=0
  g1[1] = (int)((dim0 & 0xFFFFu) << 16);                      // tensor_dim0[15:0] @63:48
  g1[2] = (int)((dim0 >> 16) | ((dim1 & 0xFFFFu) << 16));     // dim0[31:16] | dim1[15:0]
  g1[3] = (int)((dim1 >> 16) | (tile0 << 16));                // dim1[31:16] | tile_dim0
  g1[4] = (int)(tile1 & 0xFFFFu);                             // tile_dim1; tile_dim2=0
  g1[5] = (int)stride0;                                       // tensor_dim0_stride[31:0]
  g1[6] = 0;                                                  // stride hi bits, dim1_stride=0
  g1[7] = 0;
  i32x4 z4 = {0, 0, 0, 0};
#if __clang_major__ >= 23
  i32x8 z8 = {0, 0, 0, 0, 0, 0, 0, 0};
  __builtin_amdgcn_tensor_load_to_lds(g0, g1, z4, z4, z8, 0);
#else
  __builtin_amdgcn_tensor_load_to_lds(g0, g1, z4, z4, 0);
#endif
}

// ---------- WMMA GEMM: C[M,64] = A[M,K] @ B[K,64], fp32 exact ----------
// TDM stages the per-block A tile (64 x K) and the whole B (K x 64) into LDS;
// 4 waves per block, each wave owns a 16-row stripe and all 64 output columns.
__global__ void gemm_wmma_f32(const float* __restrict__ A, const float* __restrict__ B,
                              float* __restrict__ C, int M, int K) {
  __shared__ float As[64 * F_IN];   // 32 KB max (K<=128)
  __shared__ float Bs[F_IN * HID];  // 32 KB max

  const int lane = threadIdx.x & 31;
  const int hi   = lane >> 4;                   // 16-lane half
  const int ln   = lane & 15;
  const int wave = __builtin_amdgcn_readfirstlane(threadIdx.x >> 5);  // scalar wave id
  const int mblk = blockIdx.x * 64;

  if (wave == 0) {  // wave-uniform scalar branch: TDM ignores EXEC, issue once
    tdm_load_2d_f32(lds_off(As), A + (size_t)mblk * K,
                    (unsigned)K, (unsigned)(M - mblk), (unsigned)K, 64u, (unsigned)K);
    tdm_load_2d_f32(lds_off(Bs), B,
                    (unsigned)HID, (unsigned)K, (unsigned)HID, (unsigned)K, (unsigned)HID);
    __builtin_amdgcn_s_wait_tensorcnt(0);
  }
  __syncthreads();

  const int m0 = wave * 16;                     // row stripe within the block tile
  const int rowl = m0 + ln;                     // local A row for this lane

  v8f acc0 = {}, acc1 = {}, acc2 = {}, acc3 = {};
  for (int k = 0; k < K; k += 4) {
    const int ka = k + 2 * hi;
    // A frag (16x4 f32): lanes 0-15 hold K{0,1}, lanes 16-31 hold K{2,3}
    v2f a;
    a.x = As[rowl * K + ka];
    a.y = As[rowl * K + ka + 1];
    // B frags (4x16 f32) for the four 16-col tiles
    const float* B0 = Bs + ka * HID;
    const float* B1 = B0 + HID;
    v2f b0, b1, b2, b3;
    b0.x = B0[ln];      b0.y = B1[ln];
    b1.x = B0[16 + ln]; b1.y = B1[16 + ln];
    b2.x = B0[32 + ln]; b2.y = B1[32 + ln];
    b3.x = B0[48 + ln]; b3.y = B1[48 + ln];
    acc0 = __builtin_amdgcn_wmma_f32_16x16x4_f32(false, a, false, b0, (short)0, acc0, false, false);
    acc1 = __builtin_amdgcn_wmma_f32_16x16x4_f32(false, a, false, b1, (short)0, acc1, false, false);
    acc2 = __builtin_amdgcn_wmma_f32_16x16x4_f32(false, a, false, b2, (short)0, acc2, false, false);
    acc3 = __builtin_amdgcn_wmma_f32_16x16x4_f32(false, a, false, b3, (short)0, acc3, false, false);
  }
  // C layout: VGPR r, lanes 0-15 -> M=r, lanes 16-31 -> M=r+8, N=ln
#pragma unroll
  for (int r = 0; r < 8; ++r) {
    int mr = mblk + m0 + r + 8 * hi;
    if (mr < M) {
      float* o = C + (long)mr * HID;
      o[ln]      = acc0[r];
      o[16 + ln] = acc1[r];
      o[32 + ln] = acc2[r];
      o[48 + ln] = acc3[r];
    }
  }
}

// ---------- attention coefficients es/ed per (node, head) ----------
template <int H>
__global__ void attn_coeff(const float* __restrict__ h, const float* __restrict__ a_src,
                           const float* __restrict__ a_dst, float* __restrict__ es,
                           float* __restrict__ ed, int N) {
  int gid = blockIdx.x * blockDim.x + threadIdx.x;
  if (gid >= N * H) return;
  const int n = gid / H, hd = gid % H;
  constexpr int C = HID / H;
  const float* hp = h + (long)n * HID + hd * C;
  float s = 0.f, d = 0.f;
#pragma unroll
  for (int c = 0; c < C; ++c) {
    s += hp[c] * a_src[hd * C + c];
    d += hp[c] * a_dst[hd * C + c];
  }
  es[gid] = s;
  ed[gid] = d;
}

// ---------- per-edge segment max (self loops appended logically) ----------
template <int H>
__global__ void edge_max(const int* __restrict__ src, const int* __restrict__ dst,
                         const float* __restrict__ es, const float* __restrict__ ed,
                         unsigned* __restrict__ menc, int E, int N) {
  int gid = blockIdx.x * blockDim.x + threadIdx.x;
  if (gid >= (E + N) * H) return;
  const int e = gid / H, hd = gid % H;
  const int s = e < E ? src[e] : (e - E);
  const int d = e < E ? dst[e] : (e - E);
  const float v = lrelu(es[s * H + hd] + ed[d * H + hd]);
  atomicMax(&menc[d * H + hd], fenc(v));
}

// ---------- per-edge exp + segment sum ----------
template <int H>
__global__ void edge_expsum(const int* __restrict__ src, const int* __restrict__ dst,
                            const float* __restrict__ es, const float* __restrict__ ed,
                            const unsigned* __restrict__ menc, float* __restrict__ denom,
                            int E, int N) {
  int gid = blockIdx.x * blockDim.x + threadIdx.x;
  if (gid >= (E + N) * H) return;
  const int e = gid / H, hd = gid % H;
  const int s = e < E ? src[e] : (e - E);
  const int d = e < E ? dst[e] : (e - E);
  const float v = lrelu(es[s * H + hd] + ed[d * H + hd]);
  const float p = __expf(v - fdec(menc[d * H + hd]));
  atomAddF(&denom[d * H + hd], p);
}

// ---------- per-edge weighted scatter: 4 threads/edge, 16 channels each ----------
template <int H>
__global__ void edge_scatter(const int* __restrict__ src, const int* __restrict__ dst,
                             const float* __restrict__ h, const float* __restrict__ es,
                             const float* __restrict__ ed, const unsigned* __restrict__ menc,
                             const float* __restrict__ denom, float* __restrict__ out,
                             int E, int N) {
  int gid = blockIdx.x * blockDim.x + threadIdx.x;
  if (gid >= (E + N) * 4) return;
  const int e = gid >> 2, chunk = gid & 3;
  const int s = e < E ? src[e] : (e - E);
  const int d = e < E ? dst[e] : (e - E);
  constexpr int C = HID / H;
  const int hd = (chunk * 16) / C;
  const float v  = lrelu(es[s * H + hd] + ed[d * H + hd]);
  const float al = __expf(v - fdec(menc[d * H + hd])) / (denom[d * H + hd] + 1e-16f);

  const float4* hs = (const float4*)(h + (long)s * HID + chunk * 16);
  float* o = out + (long)d * HID + chunk * 16;
#pragma unroll
  for (int q = 0; q < 4; ++q) {
    float4 x = hs[q];
    atomAddF(o + 4 * q + 0, x.x * al);
    atomAddF(o + 4 * q + 1, x.y * al);
    atomAddF(o + 4 * q + 2, x.z * al);
    atomAddF(o + 4 * q + 3, x.w * al);
  }
}

// ---------- small utility kernels ----------
__global__ void fill_f32(float* p, float v, int n) {
  int i = blockIdx.x * blockDim.x + threadIdx.x; if (i < n) p[i] = v;
}
__global__ void fill_u32(unsigned* p, unsigned v, int n) {
  int i = blockIdx.x * blockDim.x + threadIdx.x; if (i < n) p[i] = v;
}
__global__ void init_bias(float* __restrict__ out, const float* __restrict__ b, int total) {
  int i = blockIdx.x * blockDim.x + threadIdx.x; if (i < total) out[i] = b[i & (HID - 1)];
}
__global__ void relu_inplace(float* p, int n) {
  int i = blockIdx.x * blockDim.x + threadIdx.x; if (i < n) p[i] = fmaxf(p[i], 0.f);
}

// ---------- pooling: per (node, channel) ----------
__global__ void pool_kernel(const float* __restrict__ h, const int* __restrict__ batch,
                            float* __restrict__ sum, unsigned* __restrict__ mx,
                            float* __restrict__ cnt, int N) {
  int gid = blockIdx.x * blockDim.x + threadIdx.x;
  if (gid >= N * HID) return;
  const int n = gid / HID, c = gid % HID;
  const int g = batch[n];
  const float v = h[gid];
  atomAddF(&sum[g * HID + c], v);
  atomicMax(&mx[g * HID + c], fenc(v));
  if (c == 0) atomAddF(&cnt[g], 1.f);
}

// ---------- head MLP: g1 = relu([mean|max|sum] @ lin1_w + lin1_b) ----------
__global__ void mlp1_kernel(const float* __restrict__ sum, const unsigned* __restrict__ mx,
                            const float* __restrict__ cnt, const float* __restrict__ w,
                            const float* __restrict__ b, float* __restrict__ g1) {
  int gid = blockIdx.x * blockDim.x + threadIdx.x;
  if (gid >= NG * HID) return;
  const int g = gid / HID, j = gid % HID;
  const float invc = 1.f / fmaxf(cnt[g], 1.f);
  float acc = b[j];
#pragma unroll 4
  for (int i = 0; i < HID; ++i) {
    const float sm = sum[g * HID + i];
    float mv = fdec(mx[g * HID + i]);
    if (!__builtin_isfinite(mv)) mv = 0.f;  // empty-graph guard (matches reference)
    acc += (sm * invc) * w[i * HID + j]
         + mv          * w[(HID + i) * HID + j]
         + sm          * w[(2 * HID + i) * HID + j];
  }
  g1[gid] = fmaxf(acc, 0.f);
}

// ---------- logits + log_softmax ----------
__global__ void logits_kernel(const float* __restrict__ g1, const float* __restrict__ w,
                              const float* __restrict__ b, float* __restrict__ out) {
  int g = blockIdx.x * blockDim.x + threadIdx.x;
  if (g >= NG) return;
  float l0 = b[0], l1 = b[1];
#pragma unroll 4
  for (int j = 0; j < HID; ++j) {
    const float v = g1[g * HID + j];
    l0 += v * w[j * NCLS + 0];
    l1 += v * w[j * NCLS + 1];
  }
  const float m = fmaxf(l0, l1);
  const float lse = m + __logf(__expf(l0 - m) + __expf(l1 - m));
  out[g * NCLS + 0] = l0 - lse;
  out[g * NCLS + 1] = l1 - lse;
}

// =====================================================================
extern "C" void kernel_launch(void* const* d_in, const int* in_sizes, int n_in,
                              void* d_out, int out_size, void* d_ws, size_t ws_size,
                              hipStream_t stream) {
  const float* x      = (const float*)d_in[0];
  const int*   ei     = (const int*)d_in[1];
  const int*   batch  = (const int*)d_in[2];
  const float* W1     = (const float*)d_in[3];
  const float* a1s    = (const float*)d_in[4];
  const float* a1d    = (const float*)d_in[5];
  const float* b1     = (const float*)d_in[6];
  const float* W2     = (const float*)d_in[7];
  const float* a2s    = (const float*)d_in[8];
  const float* a2d    = (const float*)d_in[9];
  const float* b2     = (const float*)d_in[10];
  const float* l1w    = (const float*)d_in[11];
  const float* l1b    = (const float*)d_in[12];
  const float* l2w    = (const float*)d_in[13];
  const float* l2b    = (const float*)d_in[14];
  float* out = (float*)d_out;

  const int N = in_sizes[0] / F_IN;   // 100000
  const int E = in_sizes[1] / 2;      // 3200000
  const int* src = ei;
  const int* dst = ei + E;

  // workspace carve-out
  char* ws = (char*)d_ws;
  size_t off = 0;
  auto carve = [&](size_t bytes) {
    void* p = ws + off;
    off = (off + bytes + 255) & ~(size_t)255;
    return p;
  };
  float*    bufA  = (float*)carve((size_t)N * HID * 4);   // GEMM output h
  float*    bufB  = (float*)carve((size_t)N * HID * 4);   // scatter out / next x
  float*    es    = (float*)carve((size_t)N * 4 * 4);
  float*    ed    = (float*)carve((size_t)N * 4 * 4);
  unsigned* menc  = (unsigned*)carve((size_t)N * 4 * 4);
  float*    denom = (float*)carve((size_t)N * 4 * 4);
  float*    psum  = (float*)carve((size_t)NG * HID * 4);
  unsigned* pmax  = (unsigned*)carve((size_t)NG * HID * 4);
  float*    pcnt  = (float*)carve((size_t)NG * 4);
  float*    g1    = (float*)carve((size_t)NG * HID * 4);

  const int T = 256;
  auto blks = [&](long n) { return dim3((unsigned)((n + T - 1) / T)); };
  const long EN = (long)E + N;

  // ---------------- Layer 1 (H=4) ----------------
  gemm_wmma_f32<<<dim3((N + 63) / 64), dim3(128), 0, stream>>>(x, W1, bufA, N, F_IN);
  attn_coeff<4><<<blks((long)N * 4), T, 0, stream>>>(bufA, a1s, a1d, es, ed, N);
  fill_u32<<<blks((long)N * 4), T, 0, stream>>>(menc, ENC_NEG_INF, N * 4);
  fill_f32<<<blks((long)N * 4), T, 0, stream>>>(denom, 0.f, N * 4);
  edge_max<4><<<blks(EN * 4), T, 0, stream>>>(src, dst, es, ed, menc, E, N);
  edge_expsum<4><<<blks(EN * 4), T, 0, stream>>>(src, dst, es, ed, menc, denom, E, N);
  init_bias<<<blks((long)N * HID), T, 0, stream>>>(bufB, b1, N * HID);
  edge_scatter<4><<<blks(EN * 4), T, 0, stream>>>(src, dst, bufA, es, ed, menc, denom, bufB, E, N);
  relu_inplace<<<blks((long)N * HID), T, 0, stream>>>(bufB, N * HID);

  // ---------------- Layer 2 (H=1) ----------------
  gemm_wmma_f32<<<dim3((N + 63) / 64), dim3(128), 0, stream>>>(bufB, W2, bufA, N, HID);
  attn_coeff<1><<<blks((long)N), T, 0, stream>>>(bufA, a2s, a2d, es, ed, N);
  fill_u32<<<blks((long)N), T, 0, stream>>>(menc, ENC_NEG_INF, N);
  fill_f32<<<blks((long)N), T, 0, stream>>>(denom, 0.f, N);
  edge_max<1><<<blks(EN), T, 0, stream>>>(src, dst, es, ed, menc, E, N);
  edge_expsum<1><<<blks(EN), T, 0, stream>>>(src, dst, es, ed, menc, denom, E, N);
  init_bias<<<blks((long)N * HID), T, 0, stream>>>(bufB, b2, N * HID);
  edge_scatter<1><<<blks(EN * 4), T, 0, stream>>>(src, dst, bufA, es, ed, menc, denom, bufB, E, N);

  // ---------------- Pooling + head ----------------
  fill_f32<<<blks((long)NG * HID), T, 0, stream>>>(psum, 0.f, NG * HID);
  fill_u32<<<blks((long)NG * HID), T, 0, stream>>>(pmax, ENC_NEG_INF, NG * HID);
  fill_f32<<<blks((long)NG), T, 0, stream>>>(pcnt, 0.f, NG);
  pool_kernel<<<blks((long)N * HID), T, 0, stream>>>(bufB, batch, psum, pmax, pcnt, N);
  mlp1_kernel<<<blks((long)NG * HID), T, 0, stream>>>(psum, pmax, pcnt, l1w, l1b, g1);
  logits_kernel<<<blks((long)NG), T, 0, stream>>>(g1, l2w, l2b, out);
}